// GCNConvModel_73186242724453
// MI455X (gfx1250) — compile-verified
//
#include <hip/hip_runtime.h>
#include <hip/hip_bf16.h>

typedef __attribute__((ext_vector_type(2))) float v2f;
typedef __attribute__((ext_vector_type(8))) float v8f;

// ---------------------------------------------------------------------------
// Degree / normalization
// ---------------------------------------------------------------------------
__global__ void deg_init_kernel(float* __restrict__ deg, int n) {
    int i = blockIdx.x * blockDim.x + threadIdx.x;
    if (i < n) deg[i] = 1.0f;   // self-loop contributes 1 to every node's degree
}

__global__ void deg_acc_kernel(const long long* __restrict__ dst,
                               float* __restrict__ deg, int e) {
    int i = blockIdx.x * blockDim.x + threadIdx.x;
    if (i < e) atomicAdd(&deg[(int)dst[i]], 1.0f);
}

__global__ void rsqrt_kernel(float* __restrict__ deg, int n) {
    int i = blockIdx.x * blockDim.x + threadIdx.x;
    if (i < n) deg[i] = rsqrtf(deg[i]);   // deg >= 1 always (self-loop)
}

// ---------------------------------------------------------------------------
// FP32 WMMA GEMM:  C[M x Nc] = op(A[M x K]) @ W[Nc x K]^T  (+bias) (+relu)
// One wave computes a 16 x (16*NT) strip of C via V_WMMA_F32_16X16X4_F32,
// reusing the A fragment across NT column tiles (4x A-traffic reduction).
// A-frag lane layout: lane l (half=l>>4, m=l&15): a[v] = A[row0+m][k + 2*half + v]
// B-frag lane layout:                             b[v] = W[col +m][k + 2*half + v]
//   (W row-major [out,in] => B = W^T, element (kk, n) = W[n][kk])
// C/D layout: vgpr r, lane l: element (row0 + r + 8*half, col0 + (l&15))
// ---------------------------------------------------------------------------
template <int NT, bool RELU_IN, bool RELU_OUT, bool BIAS>
__global__ __launch_bounds__(32)
void wmma_gemm_kernel(const float* __restrict__ A, const float* __restrict__ W,
                      const float* __restrict__ bias, float* __restrict__ C,
                      int K, int Nc) {
    const int row0 = blockIdx.x << 4;
    const int col0 = blockIdx.y * (NT * 16);
    const int lane = threadIdx.x;
    const int half = lane >> 4;
    const int m    = lane & 15;

    const float* Ap = A + (size_t)(row0 + m) * K + 2 * half;
    const float* Wp = W + (size_t)(col0 + m) * K + 2 * half;   // tile t adds t*16*K

    v8f acc[NT];
#pragma unroll
    for (int t = 0; t < NT; ++t) acc[t] = (v8f){0.f,0.f,0.f,0.f,0.f,0.f,0.f,0.f};

#pragma unroll 2
    for (int k = 0; k < K; k += 4) {
        v2f a = *(const v2f*)(Ap + k);
        if (RELU_IN) {
            a.x = fmaxf(a.x, 0.0f);
            a.y = fmaxf(a.y, 0.0f);
        }
#pragma unroll
        for (int t = 0; t < NT; ++t) {
            v2f b = *(const v2f*)(Wp + (size_t)t * 16 * K + k);
            acc[t] = __builtin_amdgcn_wmma_f32_16x16x4_f32(
                /*neg_a=*/false, a, /*neg_b=*/false, b,
                /*c_mod=*/(short)0, acc[t], /*reuse_a=*/false, /*reuse_b=*/false);
        }
    }

#pragma unroll
    for (int t = 0; t < NT; ++t) {
        const int colt = col0 + t * 16;
        const float bv = BIAS ? bias[colt + m] : 0.0f;
        float* Cp = C + (size_t)row0 * Nc + colt + m;
#pragma unroll
        for (int r = 0; r < 8; ++r) {
            float v = acc[t][r] + bv;
            if (RELU_OUT) v = fmaxf(v, 0.0f);
            Cp[(size_t)(r + 8 * half) * Nc] = v;
        }
    }
}

// ---------------------------------------------------------------------------
// Aggregation init: agg[i,f] = bias[f] + dis[i]^2 * t[i,f]   (self-loop term)
// ---------------------------------------------------------------------------
__global__ void agg_init_kernel(const float* __restrict__ t,
                                const float* __restrict__ dis,
                                const float* __restrict__ bias,
                                float* __restrict__ agg, int n64) {
    int i = blockIdx.x * blockDim.x + threadIdx.x;
    if (i >= n64) return;
    int node = i >> 6;
    int f    = i & 63;
    float d  = dis[node];
    agg[i] = bias[f] + d * d * t[i];
}

// ---------------------------------------------------------------------------
// Edge aggregation: one wave per edge, each lane handles 2 of 64 features.
// Feature table (25.6 MB) is L2-resident on the 192 MB L2.
// ---------------------------------------------------------------------------
__global__ void agg_edges_kernel(const long long* __restrict__ src,
                                 const long long* __restrict__ dst,
                                 const float* __restrict__ dis,
                                 const float* __restrict__ t,
                                 float* __restrict__ agg, int e) {
    int warp = (blockIdx.x * blockDim.x + threadIdx.x) >> 5;
    int lane = threadIdx.x & 31;
    if (warp >= e) return;
    int s = (int)src[warp];
    int d = (int)dst[warp];
    float w = dis[s] * dis[d];
    const float2 v = *(const float2*)(t + (size_t)s * 64 + lane * 2);
    float* op = agg + (size_t)d * 64 + lane * 2;
    atomicAdd(op,     w * v.x);
    atomicAdd(op + 1, w * v.y);
}

__global__ void sigmoid_kernel(float* __restrict__ x, int n) {
    int i = blockIdx.x * blockDim.x + threadIdx.x;
    if (i < n) x[i] = 1.0f / (1.0f + expf(-x[i]));
}

// ---------------------------------------------------------------------------
// Launcher
// ---------------------------------------------------------------------------
extern "C" void kernel_launch(void* const* d_in, const int* in_sizes, int n_in,
                              void* d_out, int out_size, void* d_ws, size_t ws_size,
                              hipStream_t stream) {
    (void)n_in; (void)out_size; (void)ws_size;

    const float*     X   = (const float*)d_in[0];      // [N,128]
    const long long* ei  = (const long long*)d_in[1];  // [2,E] int64
    const float*     W1  = (const float*)d_in[2];      // [128,128]
    const float*     B1  = (const float*)d_in[3];
    const float*     W2  = (const float*)d_in[4];      // [256,128]
    const float*     B2  = (const float*)d_in[5];
    const float*     CW1 = (const float*)d_in[6];      // [64,256]
    const float*     CB1 = (const float*)d_in[7];
    const float*     CW2 = (const float*)d_in[8];      // [64,64]
    const float*     CB2 = (const float*)d_in[9];
    float*           out = (float*)d_out;              // [N,64]

    const int N = in_sizes[0] / 128;   // 100000 (multiple of 16)
    const int E = in_sizes[1] / 2;     // 3200000
    const long long* esrc = ei;
    const long long* edst = ei + E;

    // Workspace layout (overlaid; peak ~155 MB):
    char* ws = (char*)d_ws;
    size_t off = 0;
    float* dis = (float*)(ws + off); off += (size_t)N * 4;           // deg -> dis in place
    off = (off + 255) & ~(size_t)255;
    float* h1  = (float*)(ws + off); off += (size_t)N * 128 * 4;     // fc1 out [N,128]
    float* h2  = (float*)(ws + off);                                  // fc2 out [N,256]
    float* t1   = h1;                                  // reuses h1 region (h1 dead)
    float* agg1 = h2;                                  // reuses h2 region (h2 dead)
    float* t2   = (float*)((char*)h2 + (size_t)N * 64 * 4); // second quarter of h2 region

    const int n64 = N * 64;
    const dim3 wave(32);

    // 1) degrees + normalization
    deg_init_kernel<<<(N + 255) / 256, 256, 0, stream>>>(dis, N);
    deg_acc_kernel<<<(E + 255) / 256, 256, 0, stream>>>(edst, dis, E);
    rsqrt_kernel<<<(N + 255) / 256, 256, 0, stream>>>(dis, N);

    // 2) h1 = relu(X @ W1^T + b1)      [N,128], K=128, Nc=128 -> 2 strips of 64
    wmma_gemm_kernel<4, false, true, true>
        <<<dim3(N / 16, 128 / 64), wave, 0, stream>>>(X, W1, B1, h1, 128, 128);
    // 3) h2 = h1 @ W2^T + b2           [N,256], K=128, Nc=256 -> 4 strips of 64
    wmma_gemm_kernel<4, false, false, true>
        <<<dim3(N / 16, 256 / 64), wave, 0, stream>>>(h1, W2, B2, h2, 128, 256);
    // 4) t1 = h2 @ conv1_w^T           [N,64],  K=256 (linear before aggregate)
    wmma_gemm_kernel<4, false, false, false>
        <<<dim3(N / 16, 1), wave, 0, stream>>>(h2, CW1, nullptr, t1, 256, 64);

    // 5) agg1 = conv1_b + dis^2*t1 + sum_e dis[s]dis[d]*t1[s]
    agg_init_kernel<<<(n64 + 255) / 256, 256, 0, stream>>>(t1, dis, CB1, agg1, n64);
    agg_edges_kernel<<<(E + 7) / 8, 256, 0, stream>>>(esrc, edst, dis, t1, agg1, E);

    // 6) t2 = relu(agg1) @ conv2_w^T   [N,64], K=64  (relu fused into A-load)
    wmma_gemm_kernel<4, true, false, false>
        <<<dim3(N / 16, 1), wave, 0, stream>>>(agg1, CW2, nullptr, t2, 64, 64);

    // 7) out = conv2_b + dis^2*t2 + sum_e ... ; then sigmoid in place
    agg_init_kernel<<<(n64 + 255) / 256, 256, 0, stream>>>(t2, dis, CB2, out, n64);
    agg_edges_kernel<<<(E + 7) / 8, 256, 0, stream>>>(esrc, edst, dis, t2, out, E);
    sigmoid_kernel<<<(n64 + 255) / 256, 256, 0, stream>>>(out, n64);
}